// StandardNodeParams_59734405153439
// MI455X (gfx1250) — compile-verified
//
#include <hip/hip_runtime.h>
#include <cstdint>
#include <math.h>

typedef __attribute__((ext_vector_type(2))) float v2f;
typedef __attribute__((ext_vector_type(8))) float v8f;

namespace {
constexpr int kD     = 64;
constexpr int kPairs = 4 * 2048;          // 8192 independent 64x64 matvecs
constexpr int kSlabW = kD * kD;           // W slab, row-major stride 64 dwords (identity layout
                                          // so async IOFFSET walks LDS and global in lockstep)
constexpr int kSlab  = kSlabW + kD;       // + 64 dwords for x
constexpr int kWaves = 2;                 // 2 waves/block -> 2*kSlab*4 = 33280 B LDS
constexpr float kBound    = 1.8477590650225735f;   // sqrt(2+sqrt(2))
constexpr float kInvBound = 0.5411961001461970122f;
}

// Low 32 bits of a generic pointer to LDS are the raw LDS byte offset
// (ISA 10.2: aperture lives in addr[63:32]).
__device__ __forceinline__ uint32_t lds_off(const void* p) {
  return (uint32_t)(uintptr_t)p;
}

__global__ __launch_bounds__(kWaves * 32) void
fused_norm_matvec_buntanh(const float* __restrict__ x,
                          const float* __restrict__ w,
                          const float* __restrict__ w_scale,
                          const float* __restrict__ slope,
                          float* __restrict__ out) {
  __shared__ __align__(16) float smem[kWaves * kSlab];
  const int lane = threadIdx.x & 31;
  const int wave = threadIdx.x >> 5;
  const int pair = blockIdx.x * kWaves + wave;     // exactly kPairs total

  const uint32_t ldsW = lds_off(&smem[wave * kSlab]);
  const uint32_t ldsX = ldsW + kSlabW * 4;
  const int h   = lane >> 4;     // half-wave id (0/1)
  const int c16 = lane & 15;     // column within a 16-wide group

  // ---- DMA the pair's data: x (1x B64) then W (32x B128, 2 rows each).
  // LDS layout == global layout (row-major, 256 B rows), so a single address
  // pair + IOFFSET (added to BOTH lds and global addresses) drives all 32 copies.
  {
    const uint32_t gx = (uint32_t)pair * (kD * 4) + (uint32_t)lane * 8u;
    asm volatile("global_load_async_to_lds_b64 %0, %1, %2"
                 :: "v"(ldsX + (uint32_t)lane * 8u), "v"(gx), "s"(x) : "memory");
    const uint32_t lw = ldsW + (uint32_t)h * 256u + (uint32_t)c16 * 16u;
    const uint32_t gw = (uint32_t)pair * (kD * kD * 4) + (uint32_t)h * 256u + (uint32_t)c16 * 16u;
#pragma unroll
    for (int i = 0; i < 32; ++i) {
      asm volatile("global_load_async_to_lds_b128 %0, %1, %2 offset:%3"
                   :: "v"(lw), "v"(gw), "s"(w), "i"(512 * i) : "memory");
    }
  }

  // dot accumulators: 4 column groups of 16; A-rows are all x, so every lane's
  // element 0 of C holds dot[e] for its column (M=0 row / M=8 row are identical).
  v8f acc0 = {}, acc1 = {}, acc2 = {}, acc3 = {};
  float q0 = 0.f, q1 = 0.f, q2 = 0.f, q3 = 0.f;   // per-lane partial sum-of-squares

  // K-step kk covers rows 4kk..4kk+3; this half-wave supplies rows rr=4kk+2h, rr+1.
  // A 16x4 f32 layout: lanes 0-15 hold K=0,1 / lanes 16-31 hold K=2,3 (all rows = x).
  // B 4x16 f32 layout mirrors it: V0 = row rr, V1 = row rr+1 at column g*16+c16
  //   -> exactly one ds_load_2addr_b32 (offset1 = +64 dwords = next row).
  v2f An, Bn0, Bn1, Bn2, Bn3;
  const uint32_t xaddr = ldsX + (uint32_t)h * 8u;                      // + offset:16*kk
  const uint32_t wrow0 = ldsW + (uint32_t)h * 512u + (uint32_t)c16 * 4u;

#define LOAD_SET(kk)                                                                          \
  { const uint32_t ra = wrow0 + (uint32_t)(kk) * 1024u;                                       \
    asm volatile("ds_load_b64 %0, %1 offset:%2"                                               \
                 : "=v"(An) : "v"(xaddr), "i"(16 * (kk)) : "memory");                         \
    asm volatile("ds_load_2addr_b32 %0, %1 offset0:0 offset1:64"                              \
                 : "=v"(Bn0) : "v"(ra) : "memory");                                           \
    asm volatile("ds_load_2addr_b32 %0, %1 offset0:16 offset1:80"                             \
                 : "=v"(Bn1) : "v"(ra) : "memory");                                           \
    asm volatile("ds_load_2addr_b32 %0, %1 offset0:32 offset1:96"                             \
                 : "=v"(Bn2) : "v"(ra) : "memory");                                           \
    asm volatile("ds_load_2addr_b32 %0, %1 offset0:48 offset1:112"                            \
                 : "=v"(Bn3) : "v"(ra) : "memory"); }

  // Prologue: rows [0,16) + x have landed once asynccnt <= 24 (in-order completion).
  asm volatile("s_wait_asynccnt 24" ::: "memory");
  LOAD_SET(0);

#pragma unroll
  for (int kk = 0; kk < 16; ++kk) {
    v2f Ac = An, B0 = Bn0, B1 = Bn1, B2 = Bn2, B3 = Bn3;
    if (kk < 15) {
      if (kk == 3)  asm volatile("s_wait_asynccnt 16" ::: "memory");  // rows [16,32)
      if (kk == 7)  asm volatile("s_wait_asynccnt 8"  ::: "memory");  // rows [32,48)
      if (kk == 11) asm volatile("s_wait_asynccnt 0"  ::: "memory");  // rows [48,64)
      LOAD_SET(kk + 1);
      // 5 loads of kk+1 in flight; DScnt<=5 => kk's set (issued earlier, in-order) is done.
      asm volatile("s_wait_dscnt 0x5"
                   : "+v"(Ac), "+v"(B0), "+v"(B1), "+v"(B2), "+v"(B3));
    } else {
      asm volatile("s_wait_dscnt 0x0"
                   : "+v"(Ac), "+v"(B0), "+v"(B1), "+v"(B2), "+v"(B3));
    }
    acc0 = __builtin_amdgcn_wmma_f32_16x16x4_f32(false, Ac, false, B0, (short)0, acc0, false, false);
    q0 = fmaf(B0[0], B0[0], fmaf(B0[1], B0[1], q0));
    acc1 = __builtin_amdgcn_wmma_f32_16x16x4_f32(false, Ac, false, B1, (short)0, acc1, false, false);
    q1 = fmaf(B1[0], B1[0], fmaf(B1[1], B1[1], q1));
    acc2 = __builtin_amdgcn_wmma_f32_16x16x4_f32(false, Ac, false, B2, (short)0, acc2, false, false);
    q2 = fmaf(B2[0], B2[0], fmaf(B2[1], B2[1], q2));
    acc3 = __builtin_amdgcn_wmma_f32_16x16x4_f32(false, Ac, false, B3, (short)0, acc3, false, false);
    q3 = fmaf(B3[0], B3[0], fmaf(B3[1], B3[1], q3));
  }
#undef LOAD_SET

  // ---- Epilogue: combine half-wave sq partials, normalize, bounded tanh.
  const float wsc = w_scale[0];
  const float sl  = slope[0];
  float* op = out + (size_t)pair * kD;
  float dot[4] = { acc0[0], acc1[0], acc2[0], acc3[0] };
  float qq[4]  = { q0, q1, q2, q3 };
#pragma unroll
  for (int g = 0; g < 4; ++g) {
    float s2  = qq[g] + __shfl_xor(qq[g], 16, 32);        // full column sum of squares
    float nrm = fmaxf(sqrtf(s2), 1e-12f);                 // max(norm, EPS)
    float r   = wsc * dot[g] / nrm;                       // raw_pred
    float y   = kBound * tanhf(sl * r * kInvBound);       // buntanh
    if (lane < 16) op[g * 16 + c16] = y;                  // 64B coalesced store per group
  }
}

extern "C" void kernel_launch(void* const* d_in, const int* in_sizes, int n_in,
                              void* d_out, int out_size, void* d_ws, size_t ws_size,
                              hipStream_t stream) {
  (void)in_sizes; (void)n_in; (void)out_size; (void)d_ws; (void)ws_size;
  const float* x   = (const float*)d_in[0];
  const float* w   = (const float*)d_in[1];
  const float* wsc = (const float*)d_in[2];
  const float* sl  = (const float*)d_in[3];
  // d_in[4] = step_counter: unused by the reference math
  float* out = (float*)d_out;

  dim3 grid(kPairs / kWaves);   // 4096 blocks
  dim3 block(kWaves * 32);      // 2 waves per block
  hipLaunchKernelGGL(fused_norm_matvec_buntanh, grid, block, 0, stream,
                     x, w, wsc, sl, out);
}